// DeformableAttentionSampler_47785806135299
// MI455X (gfx1250) — compile-verified
//
#include <hip/hip_runtime.h>
#include <hip/hip_bf16.h>

#define EMBED 256
#define HEADS 8
#define PTS 32
#define HD 32
#define NUM_OBJ 2048
#define NUM_IMG 8
#define IMG_H 192
#define IMG_W 256
#define FF 1024

typedef __attribute__((ext_vector_type(2))) float v2f;
typedef __attribute__((ext_vector_type(8))) float v8f;

// ---------------------------------------------------------------------------
// f32 WMMA GEMM with 2x2 register blocking:
// each wave computes a 32x32 output block as four 16x16 WMMA tiles,
// reusing each A/B fragment twice (8 loads : 4 v_wmma per K-step of 4).
// A 16x4 frag: lanes 0-15 hold M=0..15 {K0,K1}; lanes 16-31 hold {K2,K3}.
// B 4x16 frag: lanes 0-15 hold N=0..15 {K0,K1}; lanes 16-31 hold {K2,K3}.
// C/D: VGPR j, lanes 0-15 -> (M=j, N=lane); lanes 16-31 -> (M=j+8, N=lane-16).
// ---------------------------------------------------------------------------
__global__ void wmma_gemm_f32(const float* __restrict__ A,
                              const float* __restrict__ B,
                              const float* __restrict__ bias,
                              const float* __restrict__ res,
                              float* __restrict__ C,
                              int M, int N, int K, int relu)
{
    const int lane = threadIdx.x & 31;
    const int half = lane >> 4;
    const int lrow = lane & 15;
    const int n0 = blockIdx.x * 32;
    const int m0 = blockIdx.y * 32;

    v8f acc00 = {}, acc01 = {}, acc10 = {}, acc11 = {};

    const float* Arow0 = A + (size_t)(m0 + lrow) * K;
    const float* Arow1 = A + (size_t)(m0 + 16 + lrow) * K;
    const float* Bcol0 = B + (n0 + lrow);
    const float* Bcol1 = B + (n0 + 16 + lrow);

    for (int k0 = 0; k0 < K; k0 += 4) {
        const int ka = k0 + 2 * half;
        v2f a0, a1, b0, b1;
        a0.x = Arow0[ka];
        a0.y = Arow0[ka + 1];
        a1.x = Arow1[ka];
        a1.y = Arow1[ka + 1];
        b0.x = Bcol0[(size_t)ka * N];
        b0.y = Bcol0[(size_t)(ka + 1) * N];
        b1.x = Bcol1[(size_t)ka * N];
        b1.y = Bcol1[(size_t)(ka + 1) * N];
        acc00 = __builtin_amdgcn_wmma_f32_16x16x4_f32(
            false, a0, false, b0, (short)0, acc00, false, false);
        acc01 = __builtin_amdgcn_wmma_f32_16x16x4_f32(
            false, a0, false, b1, (short)0, acc01, false, false);
        acc10 = __builtin_amdgcn_wmma_f32_16x16x4_f32(
            false, a1, false, b0, (short)0, acc10, false, false);
        acc11 = __builtin_amdgcn_wmma_f32_16x16x4_f32(
            false, a1, false, b1, (short)0, acc11, false, false);
    }

    auto store_tile = [&](const v8f& acc, int mbase, int nbase) {
        const int nn = nbase + lrow;
        const float bb = bias ? bias[nn] : 0.0f;
#pragma unroll
        for (int j = 0; j < 8; ++j) {
            const int mm = mbase + j + 8 * half;
            float v = acc[j] + bb;
            if (res) v += res[(size_t)mm * N + nn];
            if (relu) v = fmaxf(v, 0.0f);
            C[(size_t)mm * N + nn] = v;
        }
    };
    store_tile(acc00, m0, n0);
    store_tile(acc01, m0, n0 + 16);
    store_tile(acc10, m0 + 16, n0);
    store_tile(acc11, m0 + 16, n0 + 16);
}

// ---------------------------------------------------------------------------
// Deformable sampling + tiny attention.
// grid = NUM_OBJ blocks of 256 threads; wave w handles head w; lanes = points.
// ---------------------------------------------------------------------------
__global__ void sample_attend(const float* __restrict__ query,
                              const float* __restrict__ key,
                              const float* __restrict__ value,
                              const float* __restrict__ img_shapes,
                              const float* __restrict__ img_transform,
                              const float* __restrict__ obj_xy,
                              const float* __restrict__ scales,
                              const int*   __restrict__ obj_img_ind,
                              const float* __restrict__ offsets,
                              float* __restrict__ vs_out,
                              float* __restrict__ mask_out,
                              float* __restrict__ x2d_out,
                              float* __restrict__ attn_out)
{
    __shared__ float lds_v[HEADS][32 * 33]; // padded tile per wave
    __shared__ float lds_a[HEADS][32];

    const int n = blockIdx.x;
    const int h = threadIdx.x >> 5;
    const int p = threadIdx.x & 31;

    const int img = obj_img_ind[n];

    // 3x3 inverse of img_transform[img] (adjugate / det)
    const float* T = img_transform + img * 9;
    const float t00 = T[0], t01 = T[1], t02 = T[2];
    const float t10 = T[3], t11 = T[4], t12 = T[5];
    const float t20 = T[6], t21 = T[7], t22 = T[8];
    const float det = t00 * (t11 * t22 - t12 * t21)
                    - t01 * (t10 * t22 - t12 * t20)
                    + t02 * (t10 * t21 - t11 * t20);
    const float id = 1.0f / det;
    const float i00 = (t11 * t22 - t12 * t21) * id;
    const float i01 = (t02 * t21 - t01 * t22) * id;
    const float i02 = (t01 * t12 - t02 * t11) * id;
    const float i10 = (t12 * t20 - t10 * t22) * id;
    const float i11 = (t00 * t22 - t02 * t20) * id;
    const float i12 = (t02 * t10 - t00 * t12) * id;
    const float i20 = (t10 * t21 - t11 * t20) * id;
    const float i21 = (t01 * t20 - t00 * t21) * id;
    const float i22 = (t00 * t11 - t01 * t10) * id;

    // sample location for this lane's point
    const float sc = scales[n];
    const float ox = offsets[(((size_t)n * HEADS + h) * PTS + p) * 2 + 0];
    const float oy = offsets[(((size_t)n * HEADS + h) * PTS + p) * 2 + 1];
    const float lx = obj_xy[n * 2 + 0] + ox * sc;
    const float ly = obj_xy[n * 2 + 1] + oy * sc;

    // projective 2d samples: x2d[i] = inv_t[i]·(lx,ly,1); out = x/z, y/z
    const float xh = i00 * lx + i01 * ly + i02;
    const float yh = i10 * lx + i11 * ly + i12;
    const float zh = i20 * lx + i21 * ly + i22;
    const size_t nh = (size_t)n * HEADS + h;
    x2d_out[(nh * 2 + 0) * PTS + p] = xh / zh;
    x2d_out[(nh * 2 + 1) * PTS + p] = yh / zh;

    // border mask (STRIDE=4, half=2; wh = flipped img_shapes)
    const float wimg = img_shapes[img * 2 + 1];
    const float himg = img_shapes[img * 2 + 0];
    const float mlx = fminf(fmaxf((lx - 2.0f) * 0.25f, 0.0f), 1.0f);
    const float mly = fminf(fmaxf((ly - 2.0f) * 0.25f, 0.0f), 1.0f);
    const float mrx = fminf(fmaxf((wimg + 2.0f - lx) * 0.25f, 0.0f), 1.0f);
    const float mry = fminf(fmaxf((himg + 2.0f - ly) * 0.25f, 0.0f), 1.0f);
    const float maskp = fminf(mlx, mrx) * fminf(mly, mry);
    mask_out[nh * PTS + p] = maskp;

    // bilinear coefficients
    const float px = lx * 0.25f - 0.5f;
    const float py = ly * 0.25f - 0.5f;
    const float xf = floorf(px);
    const float yf = floorf(py);
    const float wx = px - xf;
    const float wy = py - yf;
    const int x0 = min(max((int)xf, 0), IMG_W - 1);
    const int x1 = min(max((int)xf + 1, 0), IMG_W - 1);
    const int y0 = min(max((int)yf, 0), IMG_H - 1);
    const int y1 = min(max((int)yf + 1, 0), IMG_H - 1);
    const float w00 = (1.0f - wy) * (1.0f - wx);
    const float w01 = (1.0f - wy) * wx;
    const float w10 = wy * (1.0f - wx);
    const float w11 = wy * wx;

    const size_t HW = (size_t)IMG_H * IMG_W;
    const size_t chan_base = ((size_t)img * EMBED + (size_t)h * HD) * HW;
    const float* kbase = key + chan_base;
    const float* vbase = value + chan_base;
    const int o00 = y0 * IMG_W + x0;
    const int o01 = y0 * IMG_W + x1;
    const int o10 = y1 * IMG_W + x0;
    const int o11 = y1 * IMG_W + x1;

    const float* qb = query + nh * HD;
    float* lv = lds_v[h];
    float aacc = 0.0f;

    // gather-bound phase: unroll so the compiler can clause many independent
    // global_load_b32s per wave and keep loads in flight (latency hiding).
#pragma unroll 4
    for (int c = 0; c < HD; ++c) {
        const float* kc = kbase + (size_t)c * HW;
        const float* vc = vbase + (size_t)c * HW;
        const float k00 = kc[o00], k01 = kc[o01], k10 = kc[o10], k11 = kc[o11];
        const float v00 = vc[o00], v01 = vc[o01], v10 = vc[o10], v11 = vc[o11];
        const float ks = k00 * w00 + k01 * w01 + k10 * w10 + k11 * w11;
        const float vs = v00 * w00 + v01 * w01 + v10 * w10 + v11 * w11;
        aacc += qb[c] * ks;
        lv[c * 33 + p] = vs;
        vs_out[(nh * HD + c) * PTS + p] = vs;
    }

    // softmax over the 32 points (one wave), then mask
    aacc *= 0.17677669529663687f; // 1/sqrt(HD)
    float mx = aacc;
#pragma unroll
    for (int o = 16; o; o >>= 1) mx = fmaxf(mx, __shfl_xor(mx, o, 32));
    const float e = expf(aacc - mx);
    float ssum = e;
#pragma unroll
    for (int o = 16; o; o >>= 1) ssum += __shfl_xor(ssum, o, 32);
    lds_a[h][p] = (e / ssum) * maskp;
    __syncthreads();

    // out_h[c] = sum_p v_s[c][p] * a[p]; lanes = channels
    const int c = p;
    float acc = 0.0f;
#pragma unroll 8
    for (int pp = 0; pp < PTS; ++pp)
        acc += lds_v[h][c * 33 + pp] * lds_a[h][pp];
    attn_out[(size_t)n * EMBED + h * HD + c] = acc;
}

// ---------------------------------------------------------------------------
// Row LayerNorm over EMBED=256; one 256-thread block per row.
// ---------------------------------------------------------------------------
__global__ void layernorm_kernel(const float* __restrict__ x,
                                 const float* __restrict__ g,
                                 const float* __restrict__ b,
                                 float* __restrict__ out)
{
    __shared__ float red[8];
    const int row = blockIdx.x;
    const int t = threadIdx.x;
    const int lane = t & 31, wv = t >> 5;

    const float v = x[(size_t)row * EMBED + t];

    float s = v;
#pragma unroll
    for (int o = 16; o; o >>= 1) s += __shfl_down(s, o, 32);
    if (lane == 0) red[wv] = s;
    __syncthreads();
    float mean = 0.0f;
#pragma unroll
    for (int i = 0; i < 8; ++i) mean += red[i];
    mean *= (1.0f / 256.0f);
    __syncthreads();

    const float d = v - mean;
    float sq = d * d;
#pragma unroll
    for (int o = 16; o; o >>= 1) sq += __shfl_down(sq, o, 32);
    if (lane == 0) red[wv] = sq;
    __syncthreads();
    float var = 0.0f;
#pragma unroll
    for (int i = 0; i < 8; ++i) var += red[i];
    var *= (1.0f / 256.0f);

    out[(size_t)row * EMBED + t] = d / sqrtf(var + 1e-5f) * g[t] + b[t];
}

// ---------------------------------------------------------------------------
extern "C" void kernel_launch(void* const* d_in, const int* in_sizes, int n_in,
                              void* d_out, int out_size, void* d_ws, size_t ws_size,
                              hipStream_t stream)
{
    const float* query         = (const float*)d_in[0];
    const float* obj_emb       = (const float*)d_in[1];
    const float* key           = (const float*)d_in[2];
    const float* value         = (const float*)d_in[3];
    const float* img_shapes    = (const float*)d_in[4];
    const float* img_transform = (const float*)d_in[5];
    const float* obj_xy        = (const float*)d_in[6];
    const float* scales        = (const float*)d_in[7];
    const int*   obj_img_ind   = (const int*)d_in[8];
    const float* w_off         = (const float*)d_in[9];
    const float* b_off         = (const float*)d_in[10];
    const float* w_out         = (const float*)d_in[11];
    const float* b_out         = (const float*)d_in[12];
    const float* ln1_g         = (const float*)d_in[13];
    const float* ln1_b         = (const float*)d_in[14];
    const float* ln2_g         = (const float*)d_in[15];
    const float* ln2_b         = (const float*)d_in[16];
    const float* w_ff1         = (const float*)d_in[17];
    const float* b_ff1         = (const float*)d_in[18];
    const float* w_ff2         = (const float*)d_in[19];
    const float* b_ff2         = (const float*)d_in[20];

    // output tuple, flat in return order
    float* out      = (float*)d_out;                        // 2048*256
    float* vs_out   = out + (size_t)NUM_OBJ * EMBED;        // 2048*8*32*32
    float* mask_out = vs_out + (size_t)NUM_OBJ * HEADS * HD * PTS; // 2048*8*32
    float* x2d_out  = mask_out + (size_t)NUM_OBJ * HEADS * PTS;    // 2048*8*2*32

    // workspace partition (~20 MiB of f32)
    float* ws      = (float*)d_ws;
    float* offs    = ws;                                 // 2048*512
    float* attn    = offs + (size_t)NUM_OBJ * 512;       // 2048*256
    float* y1      = attn + (size_t)NUM_OBJ * EMBED;     // 2048*256
    float* xln     = y1   + (size_t)NUM_OBJ * EMBED;     // 2048*256
    float* hid     = xln  + (size_t)NUM_OBJ * EMBED;     // 2048*1024
    float* y2      = hid  + (size_t)NUM_OBJ * FF;        // 2048*256

    const dim3 wblk(32);

    // 1) offsets = obj_emb @ w_off + b_off      (2048x256x512)
    wmma_gemm_f32<<<dim3(512 / 32, NUM_OBJ / 32), wblk, 0, stream>>>(
        obj_emb, w_off, b_off, nullptr, offs, NUM_OBJ, 512, EMBED, 0);

    // 2) sampling + attention (gather-bound phase); also writes v_s/mask/x2d
    sample_attend<<<NUM_OBJ, 256, 0, stream>>>(
        query, key, value, img_shapes, img_transform, obj_xy, scales,
        obj_img_ind, offs, vs_out, mask_out, x2d_out, attn);

    // 3) y1 = attn @ w_out + b_out + obj_emb    (2048x256x256)
    wmma_gemm_f32<<<dim3(EMBED / 32, NUM_OBJ / 32), wblk, 0, stream>>>(
        attn, w_out, b_out, obj_emb, y1, NUM_OBJ, EMBED, EMBED, 0);

    // 4) xln = LN1(y1)
    layernorm_kernel<<<NUM_OBJ, 256, 0, stream>>>(y1, ln1_g, ln1_b, xln);

    // 5) hid = relu(xln @ w_ff1 + b_ff1)        (2048x256x1024)
    wmma_gemm_f32<<<dim3(FF / 32, NUM_OBJ / 32), wblk, 0, stream>>>(
        xln, w_ff1, b_ff1, nullptr, hid, NUM_OBJ, FF, EMBED, 1);

    // 6) y2 = hid @ w_ff2 + b_ff2 + xln         (2048x1024x256)
    wmma_gemm_f32<<<dim3(EMBED / 32, NUM_OBJ / 32), wblk, 0, stream>>>(
        hid, w_ff2, b_ff2, xln, y2, NUM_OBJ, EMBED, FF, 0);

    // 7) out = LN2(y2)
    layernorm_kernel<<<NUM_OBJ, 256, 0, stream>>>(y2, ln2_g, ln2_b, out);
}